// StructuralEncoding_45844480917699
// MI455X (gfx1250) — compile-verified
//
#include <hip/hip_runtime.h>
#include <hip/hip_bf16.h>

#define NUM_HEADS 4
#define NMAX 512
#define KP1 6   // MAX_PATH_DISTANCE + 1 -> default distance bucket

typedef __attribute__((ext_vector_type(4))) float v4f;
typedef int v4i __attribute__((vector_size(4 * sizeof(int))));   // matches builtin param
typedef __attribute__((address_space(1))) v4i* gptr_v4i;
typedef __attribute__((address_space(3))) v4i* lptr_v4i;

#if defined(__AMDGCN__) && __has_builtin(__builtin_amdgcn_global_store_async_from_lds_b128)
#define USE_ASYNC_FILL 1
#else
#define USE_ASYNC_FILL 0
#endif

// ---------------------------------------------------------------------------
// 1) Per-graph node offsets (exclusive prefix sum of num_nodes_per_graph).
//    B == 64, trivial serial scan by one lane.
// ---------------------------------------------------------------------------
__global__ void offsets_k(const int* __restrict__ nper, int* __restrict__ offs, int B) {
    if (threadIdx.x == 0) {
        int acc = 0;
        for (int b = 0; b < B; ++b) { offs[b] = acc; acc += nper[b]; }
    }
}

// ---------------------------------------------------------------------------
// 2) Bulk fill: out[b,h,i,:] = W[6,h] for all 256 MB.
// ---------------------------------------------------------------------------
#if USE_ASYNC_FILL
// Async path: stage one 512-float row per head in LDS, then stream each output
// row with 4x global_store_async_from_lds_b128 (32 lanes x 16B = 512B per op).
// One wave per output row; 8 waves / 256-thread block.
__global__ __launch_bounds__(256) void fill_async_k(const float* __restrict__ w,
                                                    float* __restrict__ out,
                                                    int nRows) {
    __shared__ float rowbuf[NUM_HEADS * NMAX];   // 8 KB
    for (int idx = threadIdx.x; idx < NUM_HEADS * NMAX; idx += blockDim.x)
        rowbuf[idx] = w[KP1 * NUM_HEADS + (idx >> 9)];   // W[6, h], h = idx/512
    __syncthreads();

    const int wave = blockIdx.x * (blockDim.x >> 5) + (threadIdx.x >> 5);
    if (wave >= nRows) return;
    const int lane = threadIdx.x & 31;
    const int h = (wave >> 9) & (NUM_HEADS - 1);     // row = ((b*H + h)*N + i)

    float* gbase = out + (size_t)wave * NMAX + lane * 4;
    const float* lbase = &rowbuf[h * NMAX + lane * 4];
#pragma unroll
    for (int k = 0; k < 4; ++k) {
        __builtin_amdgcn_global_store_async_from_lds_b128(
            (gptr_v4i)(gbase + k * 128),
            (lptr_v4i)(lbase + k * 128),
            0, 0);
    }
#if __has_builtin(__builtin_amdgcn_s_wait_asynccnt)
    __builtin_amdgcn_s_wait_asynccnt(0);
#endif
}
#else
// Fallback: one b128 non-temporal store per thread (output >> L2, stream it).
__global__ __launch_bounds__(256) void fill_nt_k(const float* __restrict__ w,
                                                 float* __restrict__ out,
                                                 size_t nQuads) {
    size_t t = (size_t)blockIdx.x * blockDim.x + threadIdx.x;
    if (t >= nQuads) return;
    // element index = t*4; row = (t*4)/512 = t>>7; h = (row>>9) & 3
    int h = (int)((t >> 16) & (NUM_HEADS - 1));
    float v = w[KP1 * NUM_HEADS + h];
    v4f val = {v, v, v, v};
    __builtin_nontemporal_store(val, (v4f*)out + t);
}
#endif

// ---------------------------------------------------------------------------
// 3) Edge scatter: out[g,h,si,di] = W[1,h]; diagonal (si==di) is skipped since
//    the diag kernel (dist 0) has final precedence.
// ---------------------------------------------------------------------------
__global__ __launch_bounds__(256) void edge_k(const int* __restrict__ ei,
                                              const int* __restrict__ batch,
                                              const int* __restrict__ offs,
                                              const float* __restrict__ w,
                                              float* __restrict__ out, int E) {
    int e = blockIdx.x * blockDim.x + threadIdx.x;
    if (e >= E) return;
    int s = ei[e];
    int d = ei[E + e];
    int g = batch[s];
    int off = offs[g];
    int si = s - off;
    int di = d - off;
    if (si == di) return;
    size_t base = (size_t)g * NUM_HEADS * NMAX * NMAX + (size_t)si * NMAX + di;
#pragma unroll
    for (int h = 0; h < NUM_HEADS; ++h)
        out[base + (size_t)h * NMAX * NMAX] = w[1 * NUM_HEADS + h];
}

// ---------------------------------------------------------------------------
// 4) Diagonal: out[b,h,i,i] = W[0,h] for i < n_b. Runs last (highest priority).
// ---------------------------------------------------------------------------
__global__ __launch_bounds__(256) void diag_k(const int* __restrict__ nper,
                                              const float* __restrict__ w,
                                              float* __restrict__ out, int B) {
    int t = blockIdx.x * blockDim.x + threadIdx.x;   // t = b*NMAX + i
    int b = t >> 9;
    int i = t & (NMAX - 1);
    if (b >= B) return;
    if (i >= nper[b]) return;
    size_t base = (size_t)b * NUM_HEADS * NMAX * NMAX + (size_t)i * NMAX + i;
#pragma unroll
    for (int h = 0; h < NUM_HEADS; ++h)
        out[base + (size_t)h * NMAX * NMAX] = w[0 * NUM_HEADS + h];
}

extern "C" void kernel_launch(void* const* d_in, const int* in_sizes, int n_in,
                              void* d_out, int out_size, void* d_ws, size_t ws_size,
                              hipStream_t stream) {
    const int*   edge_index = (const int*)d_in[0];   // [2, E] flat
    const int*   batch      = (const int*)d_in[1];   // [total_nodes]
    const int*   nper       = (const int*)d_in[2];   // [B]
    const float* w          = (const float*)d_in[3]; // [7, 4] row-major
    float*       out        = (float*)d_out;         // [B, 4, 512, 512]

    const int B = in_sizes[2];
    const int E = in_sizes[0] / 2;
    int* offs = (int*)d_ws;

    // 1) offsets
    offsets_k<<<1, 64, 0, stream>>>(nper, offs, B);

    // 2) bulk fill (256 MB streamed stores)
    const int nRows = B * NUM_HEADS * NMAX;                 // 131072 rows of 512 f32
#if USE_ASYNC_FILL
    const int wavesPerBlock = 8;                             // 256 threads, wave32
    const int fillBlocks = (nRows + wavesPerBlock - 1) / wavesPerBlock;
    fill_async_k<<<fillBlocks, 256, 0, stream>>>(w, out, nRows);
#else
    const size_t nQuads = (size_t)nRows * (NMAX / 4);        // 16,777,216 b128 stores
    const int fillBlocks = (int)((nQuads + 255) / 256);
    fill_nt_k<<<fillBlocks, 256, 0, stream>>>(w, out, nQuads);
#endif

    // 3) edges (skip self-loops), 4) diagonal last
    edge_k<<<(E + 255) / 256, 256, 0, stream>>>(edge_index, batch, offs, w, out, E);
    diag_k<<<(B * NMAX + 255) / 256, 256, 0, stream>>>(nper, w, out, B);
}